// MutateNet_52828097740994
// MI455X (gfx1250) — compile-verified
//
#include <hip/hip_runtime.h>

// MutateNet: out[n,v,l] = mask[n,l] ? onehot(cats[n,l])[v] : s[n,v,l]
// Shapes fixed by the reference: (N=256, V=4, L=100000), float32 in/out.
// Pure HBM-streaming kernel: ~1.02 GB traffic -> ~44 us at 23.3 TB/s.
// Strategy: 128-bit non-temporal loads/stores, one thread owns a 4-wide
// column group across all V=4 vocab rows so mask/cats bytes are read once.

#define NN 256
#define VV 4
#define LL 100000
#define GROUPS_PER_N (LL / 4)            // 25000 (L % 4 == 0, no tail)
#define TOTAL_GROUPS (NN * GROUPS_PER_N) // 6,400,000 threads
#define BLOCK 256                        // 8 wave32s per block

typedef __attribute__((ext_vector_type(4))) float f32x4;
typedef __attribute__((ext_vector_type(4))) int   i32x4;

__global__ __launch_bounds__(BLOCK) void mutate_kernel(
    const float* __restrict__ s,
    const int*   __restrict__ mask,   // harness marshals bool as int32
    const int*   __restrict__ cats,
    float*       __restrict__ out)
{
    const int gid = blockIdx.x * BLOCK + threadIdx.x;
    if (gid >= TOTAL_GROUPS) return;

    const int n  = gid / GROUPS_PER_N;
    const int lg = gid - n * GROUPS_PER_N;
    const int l  = lg << 2;                 // 4 consecutive l positions

    const int jl = n * LL + l;              // (n,l) flat index for mask/cats
    const int sb = n * (VV * LL) + l;       // s/out base for v = 0

    // mask/cats: each 16B chunk consumed exactly once device-wide -> NT loads.
    const i32x4 m4 = __builtin_nontemporal_load((const i32x4*)(mask + jl));
    const i32x4 c4 = __builtin_nontemporal_load((const i32x4*)(cats + jl));

#pragma unroll
    for (int v = 0; v < VV; ++v) {
        // One-shot 1 GB stream: NT on both sides keeps the 192MB L2 clean.
        const f32x4 sv = __builtin_nontemporal_load((const f32x4*)(s + sb + v * LL));
        f32x4 ov;
#pragma unroll
        for (int j = 0; j < 4; ++j) {
            const float oh = (c4[j] == v) ? 1.0f : 0.0f;   // onehot(cats)[v]
            ov[j] = m4[j] ? oh : sv[j];                    // v_cndmask select
        }
        __builtin_nontemporal_store(ov, (f32x4*)(out + sb + v * LL));
    }
}

extern "C" void kernel_launch(void* const* d_in, const int* in_sizes, int n_in,
                              void* d_out, int out_size, void* d_ws, size_t ws_size,
                              hipStream_t stream)
{
    const float* s    = (const float*)d_in[0];  // (N, V, L) float32
    const int*   mask = (const int*)  d_in[1];  // (N, L)   bool -> int32
    const int*   cats = (const int*)  d_in[2];  // (N, L)   int32
    float*       out  = (float*)d_out;          // (N, V, L) float32

    const int grid = (TOTAL_GROUPS + BLOCK - 1) / BLOCK; // 25,000 blocks
    mutate_kernel<<<grid, BLOCK, 0, stream>>>(s, mask, cats, out);
}